// Prod_cmp_68186900791724
// MI455X (gfx1250) — compile-verified
//
#include <hip/hip_runtime.h>

// Problem constants from the reference: B=256, N=65536.
#define CB 256
#define CN 65536
#define CPT 8   // complex elements per thread

// Native clang vector type: required by __builtin_nontemporal_{load,store},
// same 16B size/alignment as float4, lowers to b128 VMEM ops.
typedef __attribute__((ext_vector_type(4))) float v4f;

// Elementwise complex multiply, output layout per batch row: [reals..., imags...]
// Memory-bound: ~403 MB traffic -> ~17us floor at 23.3 TB/s. All accesses are
// 128-bit, coalesced, non-temporal (streaming; working set > 192MB L2).
__global__ __launch_bounds__(256) void Prod_cmp_kernel(
    const float* __restrict__ A,   // I1, (B, N, 2) interleaved re/im
    const float* __restrict__ Bv,  // I2, (B, N, 2) interleaved re/im
    float* __restrict__ out)       // (B, 2N): [real row | imag row] per batch
{
    const unsigned tid = blockIdx.x * blockDim.x + threadIdx.x;
    const size_t g    = (size_t)tid * CPT;   // first complex index handled
    const size_t base = g * 2;               // float offset into interleaved inputs

    const v4f* a4 = (const v4f*)(A  + base);
    const v4f* b4 = (const v4f*)(Bv + base);

    v4f a[4], b[4];
#pragma unroll
    for (int k = 0; k < 4; ++k) {
        a[k] = __builtin_nontemporal_load(a4 + k);   // global_load_b128 th:NT
        b[k] = __builtin_nontemporal_load(b4 + k);
    }

    v4f re[2], im[2];
#pragma unroll
    for (int k = 0; k < 2; ++k) {
        const v4f x0 = a[2*k], x1 = a[2*k+1];
        const v4f y0 = b[2*k], y1 = b[2*k+1];
        // 4 complex products per k (FMA-contracted by the compiler)
        re[k].x = x0.x*y0.x - x0.y*y0.y;  im[k].x = x0.y*y0.x + x0.x*y0.y;
        re[k].y = x0.z*y0.z - x0.w*y0.w;  im[k].y = x0.w*y0.z + x0.z*y0.w;
        re[k].z = x1.x*y1.x - x1.y*y1.y;  im[k].z = x1.y*y1.x + x1.x*y1.y;
        re[k].w = x1.z*y1.z - x1.w*y1.w;  im[k].w = x1.w*y1.z + x1.z*y1.w;
    }

    // Output addressing: row b has 2N floats; reals at [0,N), imags at [N,2N).
    const size_t batch = g >> 16;            // g / N   (N = 65536)
    const size_t i     = g & (CN - 1);       // g % N
    float* outr = out + (batch << 17) + i;   // batch * 2N + i
    float* outi = outr + CN;

    __builtin_nontemporal_store(re[0], (v4f*)(outr));      // global_store_b128 th:NT
    __builtin_nontemporal_store(re[1], (v4f*)(outr + 4));
    __builtin_nontemporal_store(im[0], (v4f*)(outi));
    __builtin_nontemporal_store(im[1], (v4f*)(outi + 4));
}

extern "C" void kernel_launch(void* const* d_in, const int* in_sizes, int n_in,
                              void* d_out, int out_size, void* d_ws, size_t ws_size,
                              hipStream_t stream) {
    const float* I1 = (const float*)d_in[0];   // (256, 65536, 2) f32
    const float* I2 = (const float*)d_in[1];   // (256, 65536, 2) f32
    float* out = (float*)d_out;                // (256, 65536, 2) f32 == (256, 2N) flat

    const unsigned total_threads = (CB * (unsigned)CN) / CPT;  // 2,097,152
    const unsigned block = 256;
    const unsigned grid  = total_threads / block;              // 8192

    Prod_cmp_kernel<<<grid, block, 0, stream>>>(I1, I2, out);
    (void)in_sizes; (void)n_in; (void)out_size; (void)d_ws; (void)ws_size;
}